// PlainHead_78855599555254
// MI455X (gfx1250) — compile-verified
//
#include <hip/hip_runtime.h>

typedef float v2f __attribute__((ext_vector_type(2)));
typedef float v8f __attribute__((ext_vector_type(8)));

#define B_    32
#define C_    256
#define HW_   16384     // 128*128
#define TOPK_ 1638      // int(16384 * 0.1)

// ---------------------------------------------------------------------------
// Kernel 1: scores[b, p] = sum_c x[b,c,p] * w[c] + bias, via V_WMMA_F32_16X16X4_F32.
// One wave computes 16 pixels. A-matrix = weight chunk replicated across M rows,
// B-matrix = 4 channels x 16 pixels of x. All rows of D are identical, so each
// lane's acc[0] holds the score of pixel (lane & 15).
// ---------------------------------------------------------------------------
__global__ __launch_bounds__(256) void score_wmma_kernel(
    const float* __restrict__ x, const float* __restrict__ w,
    const float* __restrict__ bias, float* __restrict__ scores) {
  const int gtid  = blockIdx.x * blockDim.x + threadIdx.x;
  const int wave  = gtid >> 5;          // global wave index
  const int lane  = threadIdx.x & 31;
  const int batch = wave >> 10;         // 1024 16-pixel tiles per image
  const int tile  = wave & 1023;
  const int pix   = (tile << 4) + (lane & 15);

  const float* xb = x + (size_t)batch * ((size_t)C_ * HW_) + pix;

  // A layout (16x4 f32): lanes 0-15 hold K=0 (V0) / K=1 (V1); lanes 16-31 hold K=2 / K=3.
  // B layout (4x16 f32): lanes 0-15 hold rows K=0 / K=1; lanes 16-31 hold rows K=2 / K=3.
  const int klo = (lane < 16) ? 0 : 2;

  v8f acc = {};
  for (int c0 = 0; c0 < C_; c0 += 4) {
    v2f a, bm;
    a.x  = w[c0 + klo];
    a.y  = w[c0 + klo + 1];
    bm.x = xb[(size_t)(c0 + klo) * HW_];
    bm.y = xb[(size_t)(c0 + klo + 1) * HW_];
    acc = __builtin_amdgcn_wmma_f32_16x16x4_f32(
        /*neg_a=*/false, a, /*neg_b=*/false, bm,
        /*c_mod=*/(short)0, acc, /*reuse_a=*/false, /*reuse_b=*/false);
  }

  const float sv = acc[0] + bias[0];   // every row of D equals score[n]
  if (lane < 16)
    scores[(size_t)batch * HW_ + (tile << 4) + lane] = sv;
}

// ---------------------------------------------------------------------------
// Kernel 2: per-batch exact top-k mean of |score| via 4-pass radix select on
// the uint bit pattern (|score| >= 0, so uint ordering == float ordering).
// One 256-thread block per batch image; data is L2-resident (2 MB total).
// ---------------------------------------------------------------------------
__global__ __launch_bounds__(256) void topk_mean_kernel(
    const float* __restrict__ scores, float* __restrict__ out) {
  __shared__ unsigned hist[256];
  __shared__ unsigned sh_prefix;
  __shared__ int      sh_need;
  __shared__ float    red[256];

  const int b   = blockIdx.x;
  const int tid = threadIdx.x;
  const float* s = scores + (size_t)b * HW_;

  if (tid == 0) { sh_prefix = 0u; sh_need = TOPK_; }
  __syncthreads();

  for (int pass = 0; pass < 4; ++pass) {
    hist[tid] = 0u;
    __syncthreads();

    const unsigned prefix = sh_prefix;
    const int shift = 24 - 8 * pass;
    for (int i = tid; i < HW_; i += 256) {
      const unsigned key = __float_as_uint(fabsf(s[i]));
      const bool match = (pass == 0) || ((key >> (shift + 8)) == prefix);
      if (match) atomicAdd(&hist[(key >> shift) & 255u], 1u);
    }
    __syncthreads();

    if (tid == 0) {
      const int need = sh_need;
      unsigned acc = 0u;
      for (int j = 255; j >= 0; --j) {
        const unsigned c = hist[j];
        if (acc + c >= (unsigned)need) {
          sh_need   = need - (int)acc;              // ties at this bucket still needed
          sh_prefix = (prefix << 8) | (unsigned)j;  // extend threshold prefix
          break;
        }
        acc += c;
      }
    }
    __syncthreads();
  }

  const unsigned T   = sh_prefix;   // exact k-th largest key (32 bits)
  const int     need = sh_need;     // how many elements equal to T to include

  float sum = 0.0f;
  for (int i = tid; i < HW_; i += 256) {
    const unsigned key = __float_as_uint(fabsf(s[i]));
    if (key > T) sum += __uint_as_float(key);
  }
  red[tid] = sum;
  __syncthreads();
  for (int off = 128; off > 0; off >>= 1) {
    if (tid < off) red[tid] += red[tid + off];
    __syncthreads();
  }
  if (tid == 0) {
    const float total = red[0] + (float)need * __uint_as_float(T);
    out[b] = total / (float)TOPK_;
  }
}

// ---------------------------------------------------------------------------
extern "C" void kernel_launch(void* const* d_in, const int* in_sizes, int n_in,
                              void* d_out, int out_size, void* d_ws, size_t ws_size,
                              hipStream_t stream) {
  (void)in_sizes; (void)n_in; (void)out_size; (void)ws_size;
  const float* x    = (const float*)d_in[0];   // [32,256,128,128] f32
  const float* w    = (const float*)d_in[1];   // [1,256,1,1]     f32
  const float* bias = (const float*)d_in[2];   // [1]             f32
  float* out    = (float*)d_out;               // [32,1]          f32
  float* scores = (float*)d_ws;                // [32,16384]      f32 (2 MB scratch)

  // 32768 waves (one per 16-pixel tile), 8 waves per 256-thread block -> 4096 blocks
  const int threads = 256;
  const int blocks  = (B_ * (HW_ / 16) * 32) / threads;
  score_wmma_kernel<<<blocks, threads, 0, stream>>>(x, w, bias, scores);
  topk_mean_kernel<<<B_, 256, 0, stream>>>(scores, out);
}